// CNN_FLOW_LAYER_34780645163522
// MI455X (gfx1250) — compile-verified
//
#include <hip/hip_runtime.h>
#include <stdint.h>

#define BATCH 16384
#define DIM   1024
#define NEG_SLOPE 0.01f
#define ROWS_PER_BLOCK 8
#define THREADS 256
#define ROW_STRIDE 1040   // 1024 data + 16 halo floats; 4160 B keeps 16B alignment per row

typedef int v4i_t __attribute__((ext_vector_type(4)));
typedef __attribute__((address_space(1))) v4i_t glob_v4i_t;
typedef __attribute__((address_space(3))) v4i_t lds_v4i_t;
typedef __attribute__((address_space(3))) float lds_float_t;

// ---- CDNA5 async global->LDS copy (16 bytes / lane) ------------------------
__device__ __forceinline__ void async_copy_b128(const float* gsrc, float* ldst) {
#if __has_builtin(__builtin_amdgcn_global_load_async_to_lds_b128)
    __builtin_amdgcn_global_load_async_to_lds_b128(
        (glob_v4i_t*)gsrc, (lds_v4i_t*)ldst, 0, 0);
#else
    uint32_t loff = (uint32_t)(uintptr_t)(lds_float_t*)ldst;
    asm volatile("global_load_async_to_lds_b128 %0, %1, off"
                 :: "v"(loff), "v"(gsrc) : "memory");
#endif
}

__device__ __forceinline__ void wait_async_zero() {
#if __has_builtin(__builtin_amdgcn_s_wait_asynccnt)
    __builtin_amdgcn_s_wait_asynccnt(0);
    asm volatile("" ::: "memory");
#else
    asm volatile("s_wait_asynccnt 0" ::: "memory");
#endif
}

// ---------------------------------------------------------------------------
__global__ __launch_bounds__(THREADS)
void cnn_flow_layer_kernel(const float* __restrict__ x,
                           const float* __restrict__ weight,
                           const float* __restrict__ bias,
                           const float* __restrict__ lmbd,
                           float* __restrict__ out,
                           float* __restrict__ logdet)
{
    __shared__ __align__(16) float s_scale[DIM];
    __shared__ __align__(16) float s_logp[DIM];   // log|scale*w0 + 1|        (conv >= 0)
    __shared__ __align__(16) float s_logn[DIM];   // log|0.01*scale*w0 + 1|   (conv <  0)
    __shared__ __align__(16) float s_x[ROWS_PER_BLOCK * ROW_STRIDE];

    const int tid  = threadIdx.x;
    const int wave = tid >> 5;
    const int lane = tid & 31;
    const int row  = blockIdx.x * ROWS_PER_BLOCK + wave;
    const size_t grow = (size_t)row * DIM;
    float* srow = &s_x[wave * ROW_STRIDE];

    // ---- phase 1: kick off async DMA of this wave's row into LDS ----------
#pragma unroll
    for (int c = 0; c < 8; ++c) {
        const int off = c * 128 + lane * 4;      // coalesced 512B per issue
        async_copy_b128(x + grow + off, srow + off);
    }
    // right-pad halo (pad = (K-1)*DILATION = 12, round to 16)
    if (lane < 16) srow[DIM + lane] = 0.0f;

    // ---- phase 2 (overlaps DMA): per-feature tables, once per block -------
    const float w0 = weight[0], w1 = weight[1], w2 = weight[2], w3 = weight[3],
                w4 = weight[4], w5 = weight[5], w6 = weight[6];
    const float b0 = bias[0];
#pragma unroll
    for (int j = 0; j < DIM / THREADS; ++j) {
        const int f = tid + j * THREADS;
        const float l  = lmbd[f];
        const float sp = fmaxf(l, 0.0f) + log1pf(expf(-fabsf(l)));  // stable softplus
        const float sc = (w0 == 0.0f) ? l
                       : ((w0 > 0.0f) ? (sp - 1.0f / w0) : (-sp - 1.0f / w0));
        s_scale[f] = sc;
        s_logp[f]  = logf(fabsf(sc * w0 + 1.0f));
        s_logn[f]  = logf(fabsf(NEG_SLOPE * sc * w0 + 1.0f));
    }
    __syncthreads();      // tables visible to all waves
    wait_async_zero();    // this wave's row is resident in LDS

    // ---- phase 3: dilated 7-tap stencil + flow math -----------------------
    float ldacc = 0.0f;
#pragma unroll
    for (int c = 0; c < 8; ++c) {
        const int i0 = c * 128 + lane * 4;
        float xa[16];
#pragma unroll
        for (int q = 0; q < 4; ++q) {           // ds_load_b128 x4: x[i0 .. i0+15]
            float4 v = *reinterpret_cast<const float4*>(srow + i0 + 4 * q);
            xa[4*q+0] = v.x; xa[4*q+1] = v.y; xa[4*q+2] = v.z; xa[4*q+3] = v.w;
        }
        const float4 sc4 = *reinterpret_cast<const float4*>(&s_scale[i0]);
        const float4 lp4 = *reinterpret_cast<const float4*>(&s_logp[i0]);
        const float4 ln4 = *reinterpret_cast<const float4*>(&s_logn[i0]);
        const float scs[4] = {sc4.x, sc4.y, sc4.z, sc4.w};
        const float lps[4] = {lp4.x, lp4.y, lp4.z, lp4.w};
        const float lns[4] = {ln4.x, ln4.y, ln4.z, ln4.w};

        float oo[4];
#pragma unroll
        for (int j = 0; j < 4; ++j) {
            float conv = b0;
            conv = fmaf(w0, xa[j +  0], conv);
            conv = fmaf(w1, xa[j +  2], conv);
            conv = fmaf(w2, xa[j +  4], conv);
            conv = fmaf(w3, xa[j +  6], conv);
            conv = fmaf(w4, xa[j +  8], conv);
            conv = fmaf(w5, xa[j + 10], conv);
            conv = fmaf(w6, xa[j + 12], conv);
            const bool pos = (conv >= 0.0f);
            const float act = pos ? conv : NEG_SLOPE * conv;
            oo[j]  = fmaf(act, scs[j], xa[j]);   // activation*scale + x (skip)
            ldacc += pos ? lps[j] : lns[j];
        }
        float4 o; o.x = oo[0]; o.y = oo[1]; o.z = oo[2]; o.w = oo[3];
        *reinterpret_cast<float4*>(out + grow + i0) = o;   // global_store_b128
    }

    // ---- wave32 reduction for logdet --------------------------------------
#pragma unroll
    for (int s = 16; s > 0; s >>= 1)
        ldacc += __shfl_down(ldacc, s, 32);
    if (lane == 0) logdet[row] = ldacc;
}

// ---------------------------------------------------------------------------
extern "C" void kernel_launch(void* const* d_in, const int* in_sizes, int n_in,
                              void* d_out, int out_size, void* d_ws, size_t ws_size,
                              hipStream_t stream) {
    (void)in_sizes; (void)n_in; (void)d_ws; (void)ws_size; (void)out_size;
    const float* x      = (const float*)d_in[0];
    const float* weight = (const float*)d_in[1];
    const float* bias   = (const float*)d_in[2];
    const float* lmbd   = (const float*)d_in[3];
    float* out    = (float*)d_out;
    float* logdet = out + (size_t)BATCH * DIM;   // outputs concatenated flat

    dim3 grid(BATCH / ROWS_PER_BLOCK);
    cnn_flow_layer_kernel<<<grid, THREADS, 0, stream>>>(x, weight, bias, lmbd,
                                                        out, logdet);
}